// Neighborhood_Computation_18090402250763
// MI455X (gfx1250) — compile-verified
//
#include <hip/hip_runtime.h>

typedef __attribute__((ext_vector_type(2))) float v2f;
typedef __attribute__((ext_vector_type(8))) float v8f;
typedef __attribute__((ext_vector_type(4))) float f4;

#define NPTS 4096
#define NB   4
#define KSEL 16
#define DAT  32

// One wave handles 32 rows (two 16x16x4 WMMA tiles). Loop over 256 column
// tiles of 16 points. S[n,m] = |c_m|^2 - 2 c_n . c_m  (row-constant |c_n|^2
// dropped: argsort-invariant).  A-row = [-2x,-2y,-2z,1], B-col = [x,y,z,|c|^2].
__global__ __launch_bounds__(256) void nbr16_kernel(const float* __restrict__ frame,
                                                    const float* __restrict__ attributes,
                                                    float* __restrict__ out)
{
    // per-wave padded transpose tile: 32 rows x 16 cols, stride 17 (bank-conflict free)
    __shared__ float lds[8][32 * 17];

    const int lane = threadIdx.x & 31;
    const int wave = threadIdx.x >> 5;
    const int b     = blockIdx.x >> 4;          // 16 blocks per batch
    const int nBase = (blockIdx.x & 15) * 256;  // 256 rows per block
    const int rowBase = nBase + wave * 32;      // 32 rows per wave
    const int lh = lane & 15;
    const bool hi = lane >= 16;

    const float* fb = frame + (size_t)b * NPTS * 12;   // frame[b,n,4,3]

    // ---- A operands (loop invariant). ISA 16x4 f32 A layout:
    // VGPR0(.x): lanes0-15 K=0, lanes16-31 K=2 ; VGPR1(.y): K=1 / K=3
    v2f a0, a1;
    {
        const float* p0 = fb + (size_t)(rowBase + lh) * 12;
        float x = p0[0], y = p0[1], z = p0[2];
        a0.x = hi ? (-2.0f * z) : (-2.0f * x);
        a0.y = hi ? 1.0f        : (-2.0f * y);
        const float* p1 = fb + (size_t)(rowBase + 16 + lh) * 12;
        x = p1[0]; y = p1[1]; z = p1[2];
        a1.x = hi ? (-2.0f * z) : (-2.0f * x);
        a1.y = hi ? 1.0f        : (-2.0f * y);
    }

    // ---- register-resident sorted top-16 (ascending; ties keep smaller index)
    float best[KSEL];
    int   bidx[KSEL];
#pragma unroll
    for (int i = 0; i < KSEL; ++i) { best[i] = 3.4e38f; bidx[i] = 0; }

    float* tile = lds[wave];
    const int m0 = hi ? 8 : 0;

    for (int ct = 0; ct < NPTS / 16; ++ct) {
        // B operand: 4x16 f32, same K-half/lane pattern as A, N = lane%16
        const float* pc = fb + (size_t)(ct * 16 + lh) * 12;
        float cx = pc[0], cy = pc[1], cz = pc[2];
        float sq = cx * cx + cy * cy + cz * cz;
        v2f bm;
        bm.x = hi ? cz : cx;
        bm.y = hi ? sq : cy;

        v8f c0 = {};
        v8f c1 = {};
        c0 = __builtin_amdgcn_wmma_f32_16x16x4_f32(false, a0, false, bm, (short)0, c0, false, false);
        c1 = __builtin_amdgcn_wmma_f32_16x16x4_f32(false, a1, false, bm, (short)0, c1, false, false);

        // transpose via LDS: C layout is VGPR v -> row v (lanes0-15) / v+8 (lanes16-31)
#pragma unroll
        for (int v = 0; v < 8; ++v) {
            tile[(v + m0) * 17 + lh]        = c0[v];
            tile[(16 + v + m0) * 17 + lh]   = c1[v];
        }

        // lane L owns combined row L; scan 16 candidates in increasing column order
#pragma unroll
        for (int j = 0; j < 16; ++j) {
            float s  = tile[lane * 17 + j];
            int   col = ct * 16 + j;
            if (s < best[KSEL - 1]) {           // strict <: stable tie-break
                best[KSEL - 1] = s;
                bidx[KSEL - 1] = col;
#pragma unroll
                for (int i = KSEL - 1; i > 0; --i) {
                    bool sw = best[i] < best[i - 1];   // strict <: equal keeps earlier idx first
                    float tv = best[i]; int ti = bidx[i];
                    best[i]     = sw ? best[i - 1] : best[i];
                    bidx[i]     = sw ? bidx[i - 1] : bidx[i];
                    best[i - 1] = sw ? tv : best[i - 1];
                    bidx[i - 1] = sw ? ti : bidx[i - 1];
                }
            }
        }
    }

    // ---- epilogue: lane owns global row n
    const int n = rowBase + lane;
    const size_t row = (size_t)b * NPTS + n;

    const float* pn0 = fb + (size_t)n * 12;
    const float cx = pn0[0], cy = pn0[1], cz = pn0[2];
    float ax[3][3];
#pragma unroll
    for (int a = 0; a < 3; ++a)
#pragma unroll
        for (int j = 0; j < 3; ++j)
            ax[a][j] = pn0[3 + a * 3 + j];

    float* outE = out + row * (KSEL * 3);                                  // (B,N,16,3)
    float* outA = out + (size_t)NB * NPTS * KSEL * 3 + row * (KSEL * DAT); // (B,N,16,32)
    const f4* attr4 = (const f4*)attributes;

#pragma unroll
    for (int k = 0; k < KSEL; ++k) {
        const int i = bidx[k];
        const float* pk = fb + (size_t)i * 12;
        float dx = pk[0] - cx, dy = pk[1] - cy, dz = pk[2] - cz;
        outE[k * 3 + 0] = dx * ax[0][0] + dy * ax[0][1] + dz * ax[0][2];
        outE[k * 3 + 1] = dx * ax[1][0] + dy * ax[1][1] + dz * ax[1][2];
        outE[k * 3 + 2] = dx * ax[2][0] + dy * ax[2][1] + dz * ax[2][2];

        const f4* src = attr4 + ((size_t)b * NPTS + i) * (DAT / 4);
        f4* dst = (f4*)(outA + (size_t)k * DAT);
#pragma unroll
        for (int q = 0; q < DAT / 4; ++q) dst[q] = src[q];
    }
}

extern "C" void kernel_launch(void* const* d_in, const int* in_sizes, int n_in,
                              void* d_out, int out_size, void* d_ws, size_t ws_size,
                              hipStream_t stream) {
    const float* frame      = (const float*)d_in[0];   // (4,4096,4,3) f32
    const float* attributes = (const float*)d_in[1];   // (4,4096,32)  f32
    float* out = (float*)d_out;                        // euclid(786432) ++ attrs(8388608)
    (void)in_sizes; (void)n_in; (void)out_size; (void)d_ws; (void)ws_size;

    // 64 blocks x 256 threads: 16 blocks/batch, 8 waves/block, 32 rows/wave
    nbr16_kernel<<<dim3(64), dim3(256), 0, stream>>>(frame, attributes, out);
}